// MultiplexRoutingHead_34565896798431
// MI455X (gfx1250) — compile-verified
//
#include <hip/hip_runtime.h>
#include <hip/hip_bf16.h>
#include <math.h>

// Problem constants (from reference)
constexpr int Pc  = 1280;
constexpr int Dc  = 512;
constexpr int Hc  = 256;
constexpr int Ec  = 4;
constexpr int Bc  = 1024;
constexpr int NFc = 4096;
constexpr int GHc = 128;
constexpr int GINc = Pc + 7;   // 1287
constexpr int Mh  = Hc / 2;    // 128

typedef float v2f __attribute__((ext_vector_type(2)));
typedef float v8f __attribute__((ext_vector_type(8)));

// ---------------- block reduction helper ----------------
template<int BS>
__device__ __forceinline__ float blk_sum(float v, float* red) {
  int t = threadIdx.x;
  red[t] = v;
  __syncthreads();
#pragma unroll
  for (int s = BS / 2; s > 0; s >>= 1) {
    if (t < s) red[t] += red[t + s];
    __syncthreads();
  }
  float r = red[0];
  __syncthreads();
  return r;
}

// ---------------- 1) per-column median + MAD (bitonic sort in LDS) ----------------
__device__ __forceinline__ void bitonic_sort_4096(float* s) {
  for (int k = 2; k <= NFc; k <<= 1) {
    for (int j = k >> 1; j > 0; j >>= 1) {
      for (int i = threadIdx.x; i < NFc; i += blockDim.x) {
        int ixj = i ^ j;
        if (ixj > i) {
          bool up = ((i & k) == 0);
          float a = s[i], b = s[ixj];
          if ((a > b) == up) { s[i] = b; s[ixj] = a; }
        }
      }
      __syncthreads();
    }
  }
}

__global__ void median_mad_kernel(const float* __restrict__ formF,
                                  const float* __restrict__ roleF,
                                  float* __restrict__ mads) {
  __shared__ float s[NFc];              // 16 KB
  int colid = blockIdx.x;               // 0 .. 2*P-1
  const float* X = (colid < Pc) ? formF : roleF;
  int col = (colid < Pc) ? colid : colid - Pc;
  for (int i = threadIdx.x; i < NFc; i += blockDim.x)
    s[i] = X[(size_t)i * Pc + col];
  __syncthreads();
  bitonic_sort_4096(s);
  float med = s[(NFc - 1) / 2];         // lower middle (torch.median semantics)
  __syncthreads();
  for (int i = threadIdx.x; i < NFc; i += blockDim.x)
    s[i] = fabsf(s[i] - med);
  __syncthreads();
  bitonic_sort_4096(s);
  if (threadIdx.x == 0) mads[colid] = s[(NFc - 1) / 2];
}

// ---------------- 2) trust vectors + gate_in assembly ----------------
__global__ void trust_gatein_kernel(const float* __restrict__ z,
                                    const float* __restrict__ fat,
                                    const float* __restrict__ rat,
                                    const float* __restrict__ jac,
                                    const float* __restrict__ mads,
                                    float* __restrict__ gate_in) {
  __shared__ float red[256];
  int t = threadIdx.x;
  float sf = 0.f, sr = 0.f;
  for (int i = t; i < Pc; i += 256) { sf += mads[i]; sr += mads[Pc + i]; }
  float ef = 0.f, er = 0.f;
  for (int i = t; i < NFc; i += 256) {
    float pf = fmaxf(fat[i], 1e-12f); ef += pf * logf(pf);
    float pr = fmaxf(rat[i], 1e-12f); er += pr * logf(pr);
  }
  sf = blk_sum<256>(sf, red);
  sr = blk_sum<256>(sr, red);
  ef = blk_sum<256>(ef, red);
  er = blk_sum<256>(er, red);
  for (int i = t; i < Pc; i += 256) gate_in[i] = z[i];
  if (t == 0) {
    float nc = log1pf((float)NFc);
    float invlog = 1.0f / logf((float)NFc);
    float entf = -ef * invlog, entr = -er * invlog;
    float madf = sf / (float)Pc, madr = sr / (float)Pc;
    gate_in[Pc + 0] = nc;
    gate_in[Pc + 1] = 1.0f - entf;
    gate_in[Pc + 2] = 1.0f / (1.0f + madf);
    gate_in[Pc + 3] = nc;
    gate_in[Pc + 4] = 1.0f - entr;
    gate_in[Pc + 5] = 1.0f / (1.0f + madr);
    gate_in[Pc + 6] = jac[0];
  }
}

// ---------------- 3) fused gate: GEMV + PReLU + LayerNorm + logits + top-2 softmax ----
__global__ void gate_kernel(const float* __restrict__ gin,
                            const float* __restrict__ gW1,
                            const float* __restrict__ gb1,
                            const float* __restrict__ ga,
                            const float* __restrict__ gg,
                            const float* __restrict__ gbeta,
                            const float* __restrict__ gW2,
                            const float* __restrict__ gb2,
                            float* __restrict__ gate_probs) {
  __shared__ float gin_s[GINc];
  __shared__ float red[GHc];
  int t = threadIdx.x;                       // 0..127
  for (int i = t; i < GINc; i += GHc) gin_s[i] = gin[i];
  __syncthreads();
  float acc = gb1[t];
  for (int i = 0; i < GINc; ++i) acc = fmaf(gin_s[i], gW1[i * GHc + t], acc);
  float a = ga[0];
  float h = (acc > 0.f) ? acc : a * acc;
  float mu = blk_sum<GHc>(h, red) * (1.0f / GHc);
  float dv = h - mu;
  float var = blk_sum<GHc>(dv * dv, red) * (1.0f / GHc);
  float hn = dv * (1.0f / sqrtf(var + 1e-5f)) * gg[t] + gbeta[t];
  float l0 = blk_sum<GHc>(hn * gW2[t * 4 + 0], red);
  float l1 = blk_sum<GHc>(hn * gW2[t * 4 + 1], red);
  float l2 = blk_sum<GHc>(hn * gW2[t * 4 + 2], red);
  float l3 = blk_sum<GHc>(hn * gW2[t * 4 + 3], red);
  if (t == 0) {
    float lg[4] = { l0 + gb2[0], l1 + gb2[1], l2 + gb2[2], l3 + gb2[3] };
    int i1 = 0;
    for (int i = 1; i < 4; ++i) if (lg[i] > lg[i1]) i1 = i;  // first occurrence wins
    int i2 = -1;
    for (int i = 0; i < 4; ++i) {
      if (i == i1) continue;
      if (i2 < 0 || lg[i] > lg[i2]) i2 = i;
    }
    float e2 = expf(lg[i2] - lg[i1]);
    float Z = 1.0f + e2;
    float pr[4] = {0.f, 0.f, 0.f, 0.f};
    pr[i1] = 1.0f / Z;
    pr[i2] = e2 / Z;
    for (int i = 0; i < 4; ++i) gate_probs[i] = pr[i];
  }
}

// ---------------- 4) p[e,h] = z . Wp[e,:,h] + bp ----------------
__global__ void p_kernel(const float* __restrict__ z,
                         const float* __restrict__ Wp,
                         const float* __restrict__ bp,
                         float* __restrict__ p) {
  int idx = blockIdx.x * 256 + threadIdx.x;   // e*H + h, 1024 total
  int e = idx >> 8, h = idx & 255;
  const float* W = Wp + (size_t)e * Pc * Hc + h;
  float acc = bp[idx];
  for (int i = 0; i < Pc; ++i) acc = fmaf(z[i], W[(size_t)i * Hc], acc);
  p[idx] = acc;
}

// ---------------- 5) t[e,o,j] = sum_i p[e,i]*Wb[e,o,i,j] ----------------------------
// Streams the 268 MB Wb tensor once (HBM-bound): float4 (b128) loads, 4 (e,o) rows per
// 256-thread block, software prefetch 16 rows ahead (global_prefetch_b8).
__global__ void t_kernel(const float* __restrict__ p,
                         const float* __restrict__ Wb,
                         float* __restrict__ tmat) {
  __shared__ float ps[Hc];
  int eo0 = blockIdx.x * 4;              // 4 consecutive (e,o) per block (same e: 256%4==0)
  int e = eo0 >> 8;
  int sub = threadIdx.x >> 6;            // 0..3 -> which (e,o)
  int jq  = threadIdx.x & 63;            // float4 column index (j = 4*jq)
  int eo = eo0 + sub;
  ps[threadIdx.x] = p[e * Hc + threadIdx.x];
  __syncthreads();
  const float4* W = (const float4*)Wb + (size_t)eo * Hc * (Hc / 4) + jq;
  float4 acc = make_float4(0.f, 0.f, 0.f, 0.f);
  for (int i = 0; i < Hc; ++i) {
    __builtin_prefetch(W + (size_t)(i + 16) * (Hc / 4), 0, 1);
    float4 w = W[(size_t)i * (Hc / 4)];
    float pv = ps[i];
    acc.x = fmaf(pv, w.x, acc.x);
    acc.y = fmaf(pv, w.y, acc.y);
    acc.z = fmaf(pv, w.z, acc.z);
    acc.w = fmaf(pv, w.w, acc.w);
  }
  ((float4*)tmat)[(size_t)eo * (Hc / 4) + jq] = acc;
}

// ---------------- 6) WMMA f32 16x16x4 GEMM, 16x64 tile/wave, fused bias (+PReLU) -----
// C[e][m,n] = sum_k A[e][m,k] * B[e][k,n] ;  BT=1 -> B stored [N,K] row-major.
// Each wave: one 16x64 C tile = 4 accumulators; A fragment reused across 4 WMMAs/k-step.
template<int BT, int ACT>
__global__ void wmma_gemm_kernel(const float* __restrict__ A,
                                 const float* __restrict__ Bm,
                                 const float* __restrict__ bias,
                                 const float* __restrict__ alpha,
                                 float* __restrict__ C,
                                 int Kd, int lda, int ldb, int ldc,
                                 long strideA, long strideB, long strideBias, long strideC,
                                 int tilesPerE, int ntilesN64) {
  int wv = (blockIdx.x * blockDim.x + threadIdx.x) >> 5;   // global wave id
  int e  = wv / tilesPerE;
  int tt = wv % tilesPerE;
  int m0 = (tt / ntilesN64) * 16;
  int n0 = (tt % ntilesN64) * 64;
  int lane = threadIdx.x & 31;
  int lo = lane & 15;
  int hi = lane >> 4;

  const float* Ar = A  + (size_t)e * strideA + (size_t)(m0 + lo) * lda;  // row of A
  const float* Be = Bm + (size_t)e * strideB;

  v8f c0 = {}, c1 = {}, c2 = {}, c3 = {};
  for (int k = 0; k < Kd; k += 4) {
    v2f av;
    av.x = Ar[k + 2 * hi];
    av.y = Ar[k + 2 * hi + 1];
    v2f b0, b1, b2, b3;
    if (BT) {
      const float* base = Be + (size_t)(n0 + lo) * ldb + k + 2 * hi;
      b0.x = base[0];                    b0.y = base[1];
      b1.x = base[(size_t)16 * ldb];     b1.y = base[(size_t)16 * ldb + 1];
      b2.x = base[(size_t)32 * ldb];     b2.y = base[(size_t)32 * ldb + 1];
      b3.x = base[(size_t)48 * ldb];     b3.y = base[(size_t)48 * ldb + 1];
    } else {
      const float* r0 = Be + (size_t)(k + 2 * hi) * ldb + n0 + lo;
      const float* r1 = Be + (size_t)(k + 1 + 2 * hi) * ldb + n0 + lo;
      b0.x = r0[0];  b0.y = r1[0];
      b1.x = r0[16]; b1.y = r1[16];
      b2.x = r0[32]; b2.y = r1[32];
      b3.x = r0[48]; b3.y = r1[48];
    }
    c0 = __builtin_amdgcn_wmma_f32_16x16x4_f32(false, av, false, b0, (short)0, c0, false, false);
    c1 = __builtin_amdgcn_wmma_f32_16x16x4_f32(false, av, false, b1, (short)0, c1, false, false);
    c2 = __builtin_amdgcn_wmma_f32_16x16x4_f32(false, av, false, b2, (short)0, c2, false, false);
    c3 = __builtin_amdgcn_wmma_f32_16x16x4_f32(false, av, false, b3, (short)0, c3, false, false);
  }

  float al = ACT ? alpha[e] : 0.0f;
  const float* biasE = bias + (size_t)e * strideBias + n0 + lo;
  float* Ce = C + (size_t)e * strideC;
  v8f cc[4] = { c0, c1, c2, c3 };
#pragma unroll
  for (int nn = 0; nn < 4; ++nn) {
    float bval = biasE[16 * nn];
#pragma unroll
    for (int v = 0; v < 8; ++v) {
      int row = m0 + v + 8 * hi;
      float val = cc[nn][v] + bval;
      if (ACT) val = (val > 0.f) ? val : al * val;
      Ce[(size_t)row * ldc + n0 + 16 * nn + lo] = val;
    }
  }
}

// ---------------- 7) scores + expert_tensor + final gated sum ----------------
__global__ void scores_kernel(const float* __restrict__ hh2,
                              const float* __restrict__ Wm2,
                              const float* __restrict__ bm2,
                              const float* __restrict__ gate_probs,
                              float* __restrict__ out) {
  int b = blockIdx.x * 256 + threadIdx.x;    // 0..1023
  float fin = 0.f;
#pragma unroll
  for (int e = 0; e < Ec; ++e) {
    const float* row = hh2 + ((size_t)e * Bc + b) * Mh;
    const float* w = Wm2 + e * Mh;
    float acc = bm2[e];
    for (int m = 0; m < Mh; ++m) acc = fmaf(row[m], w[m], acc);
    out[Bc + Ec + (size_t)b * Ec + e] = acc;       // expert_tensor [B,E]
    fin += acc * gate_probs[e];
  }
  out[b] = fin;                                    // final_scores [B]
}

// ---------------- launcher ----------------
extern "C" void kernel_launch(void* const* d_in, const int* in_sizes, int n_in,
                              void* d_out, int out_size, void* d_ws, size_t ws_size,
                              hipStream_t stream) {
  (void)in_sizes; (void)n_in; (void)out_size; (void)ws_size;
  const float* z     = (const float*)d_in[0];
  const float* formF = (const float*)d_in[1];
  const float* roleF = (const float*)d_in[2];
  const float* q     = (const float*)d_in[3];
  const float* fat   = (const float*)d_in[4];
  const float* rat   = (const float*)d_in[5];
  const float* jac   = (const float*)d_in[6];
  const float* gW1   = (const float*)d_in[7];
  const float* gb1   = (const float*)d_in[8];
  const float* ga    = (const float*)d_in[9];
  const float* gg    = (const float*)d_in[10];
  const float* gbeta = (const float*)d_in[11];
  const float* gW2   = (const float*)d_in[12];
  const float* gb2   = (const float*)d_in[13];
  const float* Wp    = (const float*)d_in[14];
  const float* bp    = (const float*)d_in[15];
  const float* Wd    = (const float*)d_in[16];
  const float* bd    = (const float*)d_in[17];
  const float* Wb    = (const float*)d_in[18];
  const float* bb    = (const float*)d_in[19];
  const float* a1    = (const float*)d_in[20];
  const float* Wm1   = (const float*)d_in[21];
  const float* bm1   = (const float*)d_in[22];
  const float* a2    = (const float*)d_in[23];
  const float* Wm2   = (const float*)d_in[24];
  const float* bm2   = (const float*)d_in[25];

  float* ws      = (float*)d_ws;
  float* mads    = ws;                        // 2*P = 2560
  float* gate_in = ws + 2560;                 // 1287
  float* pbuf    = ws + 4096;                 // E*H = 1024
  float* dbuf    = ws + 8192;                 // E*B*H = 1048576
  float* tbuf    = dbuf + (size_t)Ec * Bc * Hc;        // E*H*H = 262144
  float* jbuf    = tbuf + (size_t)Ec * Hc * Hc;        // E*B*H = 1048576
  float* h2buf   = jbuf + (size_t)Ec * Bc * Hc;        // E*B*128 = 524288
  float* out     = (float*)d_out;             // [final(1024) | gate_probs(4) | ET(4096)]

  // 1) medians/MADs: one workgroup per column, bitonic sort in LDS
  median_mad_kernel<<<2 * Pc, 256, 0, stream>>>(formF, roleF, mads);
  // 2) trust vectors + gate_in
  trust_gatein_kernel<<<1, 256, 0, stream>>>(z, fat, rat, jac, mads, gate_in);
  // 3) fused gate -> gate_probs at out[B..B+3]
  gate_kernel<<<1, GHc, 0, stream>>>(gate_in, gW1, gb1, ga, gg, gbeta, gW2, gb2, out + Bc);
  // 4) p = z @ Wp + bp
  p_kernel<<<(Ec * Hc) / 256, 256, 0, stream>>>(z, Wp, bp, pbuf);
  // 5) d[e] = q @ Wd[e] + bd[e]   (WMMA, M=1024 N=256 K=512, 16x64 tiles)
  {
    int tilesPerE = (Bc / 16) * (Hc / 64);            // 256
    wmma_gemm_kernel<0, 0><<<(Ec * tilesPerE) / 4, 128, 0, stream>>>(
        q, Wd, bd, nullptr, dbuf,
        Dc, Dc, Hc, Hc,
        0L, (long)Dc * Hc, (long)Hc, (long)Bc * Hc,
        tilesPerE, Hc / 64);
  }
  // 6) t[e,o,:] = p[e] @ Wb[e,o]   (streams 268 MB once -> HBM bound)
  t_kernel<<<(Ec * Hc) / 4, 256, 0, stream>>>(pbuf, Wb, tbuf);
  // 7) joint[e] = d[e] @ t[e]^T + bb[e], PReLU(a1)   (WMMA, B stored [N,K])
  {
    int tilesPerE = (Bc / 16) * (Hc / 64);            // 256
    wmma_gemm_kernel<1, 1><<<(Ec * tilesPerE) / 4, 128, 0, stream>>>(
        dbuf, tbuf, bb, a1, jbuf,
        Hc, Hc, Hc, Hc,
        (long)Bc * Hc, (long)Hc * Hc, (long)Hc, (long)Bc * Hc,
        tilesPerE, Hc / 64);
  }
  // 8) hh2[e] = joint[e] @ Wm1[e] + bm1[e], PReLU(a2)   (WMMA, N=128)
  {
    int tilesPerE = (Bc / 16) * (Mh / 64);            // 128
    wmma_gemm_kernel<0, 1><<<(Ec * tilesPerE) / 4, 128, 0, stream>>>(
        jbuf, Wm1, bm1, a2, h2buf,
        Hc, Hc, Mh, Mh,
        (long)Bc * Hc, (long)Hc * Mh, (long)Mh, (long)Bc * Mh,
        tilesPerE, Mh / 64);
  }
  // 9) scores -> expert_tensor, final gated sum
  scores_kernel<<<Bc / 256, 256, 0, stream>>>(h2buf, Wm2, bm2, out + Bc, out);
}